// Encoder_7756710936715
// MI455X (gfx1250) — compile-verified
//
#include <hip/hip_runtime.h>

// ---------------------------------------------------------------------------
// Sparse 3D CNN encoder (Minkowski-style) for MI455X / gfx1250.
// All 12 convolutions run as implicit GEMM on V_WMMA_F32_16X16X4_F32
// (full-fp32 matrix pipe -> matches fp32 reference numerics).
// Power-of-two dims -> shifts/masks only; no integer division in hot loops.
// Pair-interleaved weights -> A and B fragments are single b64 loads.
// Branchless bounds handling (cndmask on 64-bit offset) -> WMMA always runs
// with EXEC all-ones. MT=2 x NT=2 register blocking -> 1 load per WMMA.
// ---------------------------------------------------------------------------

typedef __attribute__((ext_vector_type(2))) float v2f;
typedef __attribute__((ext_vector_type(8))) float v8f;

#define NBATCH 4

__device__ __forceinline__ float eluf(float t) { return t > 0.f ? t : expm1f(t); }

// ---------------------------------------------------------------------------
// Generic conv 3x3x3 (pad=1), Cin multiple of 16, channels-last.
//   in  : [N, Di^3, Cin]   wp2 : pair-interleaved [K/2][Cout][2], K=tap*Cin+ci
//   out : [N*Do^3, Cout]   mlev: output-level mask, fused multiply
// One wave: (MT*16) voxels x (NT*16) output channels.
// ---------------------------------------------------------------------------
template <int MT, int NT>
__global__ void __launch_bounds__(256)
conv3d_wmma_cx(const float* __restrict__ in, const float* __restrict__ wp2,
               const float* __restrict__ mlev, float* __restrict__ out,
               int diLog2, int doLog2, int cinLog2, int coutLog2,
               int coTileLog2, int stride)
{
    const int lane    = threadIdx.x & 31;
    const int waveId  = blockIdx.x * 8 + (threadIdx.x >> 5);
    const int half    = lane >> 4;        // K pair (0,1) vs (2,3)
    const int ln      = lane & 15;
    const int voxTile = waveId >> coTileLog2;         // MT*16 voxels
    const int co0     = (waveId & ((1 << coTileLog2) - 1)) * (16 * NT);
    const int Di  = 1 << diLog2;
    const int Do  = 1 << doLog2;
    const int Cin = 1 << cinLog2;

    // ---- per-M-tile precompute: base ptr, per-axis offsets + validity ------
    const float* base[MT];
    long zoff[MT][3], yoff[MT][3], xoff[MT][3];
    int  zok[MT][3],  yok[MT][3],  xok[MT][3];
#pragma unroll
    for (int mq = 0; mq < MT; ++mq) {
        const int vA = voxTile * (MT * 16) + mq * 16 + ln;
        const int xo = vA & (Do - 1);
        const int yo = (vA >> doLog2) & (Do - 1);
        const int zo = (vA >> (2 * doLog2)) & (Do - 1);
        const int n  = vA >> (3 * doLog2);
        const int xb = xo * stride - 1, yb = yo * stride - 1, zb = zo * stride - 1;
        base[mq] = in + ((long)n << (3 * diLog2 + cinLog2)) + (half << 1);
#pragma unroll
        for (int d = 0; d < 3; ++d) {
            const int zi = zb + d, yi = yb + d, xi = xb + d;
            zok[mq][d] = (unsigned)zi < (unsigned)Di;
            yok[mq][d] = (unsigned)yi < (unsigned)Di;
            xok[mq][d] = (unsigned)xi < (unsigned)Di;
            zoff[mq][d] = (long)zi << (2 * diLog2 + cinLog2);
            yoff[mq][d] = (long)yi << (diLog2 + cinLog2);
            xoff[mq][d] = (long)xi << cinLog2;
        }
    }

    v8f acc[MT][NT] = {};

    // B pointer walks linearly through all taps x channels (pair layout)
    const float* pb = wp2 + ((long)half << (coutLog2 + 1)) + ((co0 + ln) << 1);

#pragma unroll
    for (int dz = 0; dz < 3; ++dz)
#pragma unroll
        for (int dy = 0; dy < 3; ++dy)
#pragma unroll
            for (int dx = 0; dx < 3; ++dx) {
                long aoff[MT];
                int  okm[MT];
#pragma unroll
                for (int mq = 0; mq < MT; ++mq) {
                    okm[mq]  = zok[mq][dz] & yok[mq][dy] & xok[mq][dx];
                    long o   = zoff[mq][dz] + yoff[mq][dy] + xoff[mq][dx];
                    aoff[mq] = okm[mq] ? o : 0;   // data-flow select, no branch
                }
#pragma unroll 2
                for (int ci0 = 0; ci0 < Cin; ci0 += 4) {
                    v2f a[MT];
#pragma unroll
                    for (int mq = 0; mq < MT; ++mq) {
                        v2f t = *(const v2f*)(base[mq] + aoff[mq] + ci0);
                        a[mq].x = okm[mq] ? t.x : 0.f;
                        a[mq].y = okm[mq] ? t.y : 0.f;
                    }
#pragma unroll
                    for (int q = 0; q < NT; ++q) {
                        v2f b = *(const v2f*)(pb + (q << 5));   // global_load_b64
#pragma unroll
                        for (int mq = 0; mq < MT; ++mq)
                            acc[mq][q] = __builtin_amdgcn_wmma_f32_16x16x4_f32(
                                false, a[mq], false, b, (short)0, acc[mq][q],
                                false, false);
                    }
                    pb += (4 << coutLog2);        // advance 4 K values
                }
            }

    // C/D layout: VGPR r -> M = r + 8*half, N = ln. Fused output-mask.
#pragma unroll
    for (int mq = 0; mq < MT; ++mq)
#pragma unroll
        for (int r = 0; r < 8; ++r) {
            const int v = voxTile * (MT * 16) + mq * 16 + r + (half << 3);
            const float m = mlev[v];
#pragma unroll
            for (int q = 0; q < NT; ++q)
                out[((long)v << coutLog2) + co0 + q * 16 + ln] = acc[mq][q][r] * m;
        }
}

// ---------------------------------------------------------------------------
// Layer-0 conv: Cin=1, Cout=16, Di=128, Do=64, stride=2 (all hardcoded).
// K = tap in [0,27), padded to 28. Fully unrolled: each lane's two candidate
// taps per step are compile-time constants selected by `half` (no division).
// ---------------------------------------------------------------------------
__global__ void __launch_bounds__(256)
conv3d_wmma_c1(const float* __restrict__ in, const float* __restrict__ wp2,
               const float* __restrict__ mlev, float* __restrict__ out)
{
    const int lane   = threadIdx.x & 31;
    const int waveId = blockIdx.x * 8 + (threadIdx.x >> 5);
    const int half   = lane >> 4;
    const int ln     = lane & 15;

    const int vA = (waveId << 4) + ln;
    const int xo = vA & 63, yo = (vA >> 6) & 63, zo = (vA >> 12) & 63, n = vA >> 18;
    const int xb = 2 * xo - 1, yb = 2 * yo - 1, zb = 2 * zo - 1;
    const float* base = in + ((long)n << 21);    // 128^3

    int zoffs[3], yoffs[3], xoffs[3], zokA[3], yokA[3], xokA[3];
#pragma unroll
    for (int d = 0; d < 3; ++d) {
        const int zi = zb + d, yi = yb + d, xi = xb + d;
        zokA[d] = (unsigned)zi < 128u; zoffs[d] = zi << 14;
        yokA[d] = (unsigned)yi < 128u; yoffs[d] = yi << 7;
        xokA[d] = (unsigned)xi < 128u; xoffs[d] = xi;
    }

    v8f acc = {};
    const float* pb = wp2 + (half << 5) + (ln << 1);  // pair row, Cout=16

#pragma unroll
    for (int k0 = 0; k0 < 28; k0 += 4) {
        v2f a;
#pragma unroll
        for (int u = 0; u < 2; ++u) {
            // candidate taps (compile-time constants after unroll)
            const int Kl = k0 + u,     Kh = k0 + 2 + u;
            const int dzl = Kl / 9, dyl = (Kl % 9) / 3, dxl = Kl % 3;
            const int dzh = (Kh < 27) ? Kh / 9 : 0;
            const int dyh = (Kh < 27) ? (Kh % 9) / 3 : 0;
            const int dxh = (Kh < 27) ? Kh % 3 : 0;
            const int kvl = Kl < 27, kvh = Kh < 27;
            // per-lane selects of precomputed constants (cndmask, no div)
            const int zo_ = half ? zoffs[dzh] : zoffs[dzl];
            const int yo_ = half ? yoffs[dyh] : yoffs[dyl];
            const int xo_ = half ? xoffs[dxh] : xoffs[dxl];
            const int ok  = (half ? (kvh & zokA[dzh] & yokA[dyh] & xokA[dxh])
                                  : (kvl & zokA[dzl] & yokA[dyl] & xokA[dxl]));
            const long off = ok ? (long)(zo_ + yo_ + xo_) : 0;
            float av = base[off];
            av = ok ? av : 0.f;
            if (u == 0) a.x = av; else a.y = av;
        }
        v2f b = *(const v2f*)pb;       // K=27 row zero-padded in wp2
        pb += 64;                      // 2 pairs * 32
        acc = __builtin_amdgcn_wmma_f32_16x16x4_f32(false, a, false, b,
                                                    (short)0, acc, false, false);
    }

#pragma unroll
    for (int r = 0; r < 8; ++r) {
        const int v = (waveId << 4) + r + (half << 3);
        out[((long)v << 4) + ln] = acc[r] * mlev[v];
    }
}

// ---- weight repack: OIDHW -> pair-interleaved [K/2][Cout][2] ---------------
__global__ void __launch_bounds__(256)
repack_pairs(const float* __restrict__ w, float* __restrict__ wp2,
             int cinLog2, int coutLog2, int KtotPad)
{
    const int idx = blockIdx.x * blockDim.x + threadIdx.x;
    const int Cout = 1 << coutLog2;
    if (idx >= (KtotPad << coutLog2)) return;
    const int co = idx & (Cout - 1);
    const int K  = idx >> coutLog2;
    const int Ktot = 27 << cinLog2;
    float val = 0.f;
    if (K < Ktot) {
        const int ci  = K & ((1 << cinLog2) - 1);
        const int tap = K >> cinLog2;
        val = w[(((long)co << cinLog2) + ci) * 27 + tap];
    }
    wp2[((long)(K >> 1) << (coutLog2 + 1)) + (co << 1) + (K & 1)] = val;
}

// ---- mask 2x2x2 max-pool (stride 2) ---------------------------------------
__global__ void __launch_bounds__(256)
pool_mask_u8(const unsigned char* __restrict__ mi, float* __restrict__ mo, int Do)
{
    int idx = blockIdx.x * blockDim.x + threadIdx.x;
    int tot = NBATCH * Do * Do * Do;
    if (idx >= tot) return;
    const int Di = Do * 2;
    int x = idx % Do; int t = idx / Do;
    int y = t % Do;   t /= Do;
    int z = t % Do;   int n = t / Do;
    long base = (((long)n * Di + 2 * z) * Di + 2 * y) * Di + 2 * x;
    unsigned acc = 0;
    for (int dz = 0; dz < 2; ++dz)
        for (int dy = 0; dy < 2; ++dy)
            for (int dx = 0; dx < 2; ++dx)
                acc |= mi[base + ((long)dz * Di + dy) * Di + dx];
    mo[idx] = acc ? 1.f : 0.f;
}

__global__ void __launch_bounds__(256)
pool_mask_f(const float* __restrict__ mi, float* __restrict__ mo, int Do)
{
    int idx = blockIdx.x * blockDim.x + threadIdx.x;
    int tot = NBATCH * Do * Do * Do;
    if (idx >= tot) return;
    const int Di = Do * 2;
    int x = idx % Do; int t = idx / Do;
    int y = t % Do;   t /= Do;
    int z = t % Do;   int n = t / Do;
    long base = (((long)n * Di + 2 * z) * Di + 2 * y) * Di + 2 * x;
    float acc = 0.f;
    for (int dz = 0; dz < 2; ++dz)
        for (int dy = 0; dy < 2; ++dy)
            for (int dx = 0; dx < 2; ++dx)
                acc = fmaxf(acc, mi[base + ((long)dz * Di + dy) * Di + dx]);
    mo[idx] = acc;
}

// ---- deterministic single-block sum (active-site count per level) ----------
__global__ void __launch_bounds__(256)
reduce_sum1(const float* __restrict__ v, int n, float* __restrict__ outp)
{
    __shared__ float sm[256];
    float s = 0.f;
    for (int i = threadIdx.x; i < n; i += 256) s += v[i];
    sm[threadIdx.x] = s;
    __syncthreads();
    for (int w = 128; w > 0; w >>= 1) {
        if ((int)threadIdx.x < w) sm[threadIdx.x] += sm[threadIdx.x + w];
        __syncthreads();
    }
    if (threadIdx.x == 0) outp[0] = sm[0];
}

// ---- BN statistics stage 1: 16 channels x 16 voxels per block (coalesced) --
__global__ void __launch_bounds__(256)
bn_stats_partial(const float* __restrict__ y, int V, int cLog2,
                 float* __restrict__ part)
{
    const int C  = 1 << cLog2;
    const int cl = threadIdx.x & 15;
    const int vt = threadIdx.x >> 4;                 // 0..15
    const int c  = (blockIdx.y << 4) + cl;
    const int S  = gridDim.x;
    float s = 0.f, s2 = 0.f;
    for (int v = blockIdx.x * 16 + vt; v < V; v += S * 16) {
        float t = y[((long)v << cLog2) + c];
        s += t; s2 += t * t;
    }
    __shared__ float sa[16][17], sb[16][17];
    sa[vt][cl] = s; sb[vt][cl] = s2;
    __syncthreads();
    for (int w = 8; w > 0; w >>= 1) {
        if (vt < w) { sa[vt][cl] += sa[vt + w][cl]; sb[vt][cl] += sb[vt + w][cl]; }
        __syncthreads();
    }
    if (vt == 0) {
        part[(long)c * S + blockIdx.x]       = sa[0][cl];
        part[(long)(C + c) * S + blockIdx.x] = sb[0][cl];
    }
}

// ---- BN statistics stage 2 -------------------------------------------------
__global__ void __launch_bounds__(256)
bn_stats_final(const float* __restrict__ part, int S, int C, float* __restrict__ stats)
{
    int i = blockIdx.x * blockDim.x + threadIdx.x;
    if (i >= 2 * C) return;
    float s = 0.f;
    for (int j = 0; j < S; ++j) s += part[(long)i * S + j];
    stats[i] = s;
}

// ---- fused BN (active-site moments) + ELU + re-mask, in place --------------
__global__ void __launch_bounds__(256)
bn_apply_elu(float* __restrict__ y, const float* __restrict__ mlev,
             const float* __restrict__ stats, const float* __restrict__ cnt,
             const float* __restrict__ gamma, const float* __restrict__ beta,
             long tot, int cLog2)
{
    long idx = (long)blockIdx.x * blockDim.x + threadIdx.x;
    if (idx >= tot) return;
    const int  C = 1 << cLog2;
    const int  c = (int)(idx & (C - 1));
    const long v = idx >> cLog2;
    const float nact = cnt[0];
    const float mu   = stats[c] / nact;
    float var = stats[C + c] / nact - mu * mu;
    var = var > 0.f ? var : 0.f;
    const float inv = rsqrtf(var + 1e-5f);
    const float tv  = (y[idx] - mu) * inv * gamma[c] + beta[c];
    y[idx] = (mlev[v] != 0.f) ? eluf(tv) : 0.f;     // mask before ELU (elu(0)=0)
}

// ---- final: per-batch masked spatial sum at level 6 (2^3, C=512) -----------
__global__ void __launch_bounds__(512)
final_reduce(const float* __restrict__ y6, const float* __restrict__ m6,
             float* __restrict__ S, float* __restrict__ nb)
{
    const int b = blockIdx.x, ci = threadIdx.x;     // 4 blocks x 512 threads
    float s = 0.f;
    for (int v = 0; v < 8; ++v) s += y6[((b * 8 + v) * 512) + ci];
    S[b * 512 + ci] = s;
    if (ci == 0) {
        float nn = 0.f;
        for (int v = 0; v < 8; ++v) nn += m6[b * 8 + v];
        nb[b] = nn;
    }
}

// out[b,co] = (1/nb[b]) * sum_ci Wf[co,ci] * S[b,ci]   (1x1 conv + global avg)
__global__ void __launch_bounds__(256)
final_gemm(const float* __restrict__ S, const float* __restrict__ wf,
           const float* __restrict__ nb, float* __restrict__ out)
{
    int idx = blockIdx.x * blockDim.x + threadIdx.x;
    if (idx >= 4 * 1024) return;
    const int b = idx >> 10, co = idx & 1023;
    float s = 0.f;
    for (int ci = 0; ci < 512; ++ci) s += wf[co * 512 + ci] * S[b * 512 + ci];
    out[idx] = s / nb[b];
}

// ---------------------------------------------------------------------------
static inline int ilog2i(int v) { int l = 0; while ((1 << l) < v) ++l; return l; }

extern "C" void kernel_launch(void* const* d_in, const int* in_sizes, int n_in,
                              void* d_out, int out_size, void* d_ws, size_t ws_size,
                              hipStream_t stream)
{
    (void)in_sizes; (void)n_in; (void)out_size; (void)ws_size;

    static const int CH[6] = {16, 32, 64, 128, 256, 512};
    static const int DO[6] = {64, 32, 16, 8, 4, 2};

    const float*         x     = (const float*)d_in[0];
    const unsigned char* mask0 = (const unsigned char*)d_in[1];  // jnp bool -> 1B
    const float*         wfin  = (const float*)d_in[2 + 6 * 6];  // d_in[38]

    // -------- workspace carve-up (deterministic offsets) --------------------
    char*  ws  = (char*)d_ws;
    size_t off = 0;
    auto carve = [&](size_t bytes) -> void* {
        void* p = ws + off;
        off += (bytes + 255) & ~(size_t)255;
        return p;
    };
    float* wp2 = (float*)carve((size_t)28 * 512 * 512 * 4);   // max padded repack
    float* mlev[6];
    for (int l = 0; l < 6; ++l)
        mlev[l] = (float*)carve((size_t)NBATCH * DO[l] * DO[l] * DO[l] * 4);
    float* cnt   = (float*)carve(6 * 4);
    float* stats = (float*)carve(2 * 512 * 4);
    float* part  = (float*)carve((size_t)2 * 512 * 64 * 4);
    float* Sbuf  = (float*)carve(4 * 512 * 4);
    float* nb    = (float*)carve(4 * 4);
    const size_t actElems = (size_t)NBATCH * 64 * 64 * 64 * 16;  // largest level
    float* bufA = (float*)carve(actElems * 4);
    float* bufB = (float*)carve(actElems * 4);

    // -------- mask pyramid + active counts ----------------------------------
    {
        int tot = NBATCH * 64 * 64 * 64;
        pool_mask_u8<<<(tot + 255) / 256, 256, 0, stream>>>(mask0, mlev[0], 64);
        for (int l = 1; l < 6; ++l) {
            tot = NBATCH * DO[l] * DO[l] * DO[l];
            pool_mask_f<<<(tot + 255) / 256, 256, 0, stream>>>(mlev[l - 1], mlev[l], DO[l]);
        }
        for (int l = 0; l < 6; ++l) {
            tot = NBATCH * DO[l] * DO[l] * DO[l];
            reduce_sum1<<<1, 256, 0, stream>>>(mlev[l], tot, cnt + l);
        }
    }

    // -------- encoder blocks ------------------------------------------------
    const float* src = x;   // level0: NCDHW with Cin=1 == channels-last
    int Cin = 1;
    for (int i = 0; i < 6; ++i) {
        const int C    = CH[i];
        const int DiL  = 128 >> i;
        const int DoL  = DiL >> 1;
        const int V    = NBATCH * DoL * DoL * DoL;
        const int diL2 = ilog2i(DiL), doL2 = ilog2i(DoL);
        const int ciL2 = ilog2i(Cin), coL2 = ilog2i(C);

        const float* wa = (const float*)d_in[2 + 6 * i + 0];
        const float* wb = (const float*)d_in[2 + 6 * i + 1];
        const float* ga = (const float*)d_in[2 + 6 * i + 2];
        const float* ba = (const float*)d_in[2 + 6 * i + 3];
        const float* gb = (const float*)d_in[2 + 6 * i + 4];
        const float* bb = (const float*)d_in[2 + 6 * i + 5];

        // ---- stride-2 conv ----
        {
            const int KtotPad = (27 * Cin + 1) & ~1;
            repack_pairs<<<((KtotPad << coL2) + 255) / 256, 256, 0, stream>>>(
                wa, wp2, ciL2, coL2, KtotPad);
            if (i == 0) {
                conv3d_wmma_c1<<<65536 / 8, 256, 0, stream>>>(src, wp2, mlev[0], bufA);
            } else {
                const int ctL2  = coL2 - 5;                 // log2(Cout/32), NT=2
                const int waves = (V / 32) << ctL2;         // MT=2
                conv3d_wmma_cx<2, 2><<<waves / 8, 256, 0, stream>>>(
                    src, wp2, mlev[i], bufA, diL2, doL2, ciL2, coL2, ctL2, 2);
            }
        }
        // ---- BN + ELU ----
        dim3 gs(64, C / 16);
        bn_stats_partial<<<gs, 256, 0, stream>>>(bufA, V, coL2, part);
        bn_stats_final<<<(2 * C + 255) / 256, 256, 0, stream>>>(part, 64, C, stats);
        const long tot = (long)V * C;
        bn_apply_elu<<<(int)((tot + 255) / 256), 256, 0, stream>>>(
            bufA, mlev[i], stats, cnt + i, ga, ba, tot, coL2);

        // ---- stride-1 conv ----
        {
            const int KtotPad = 27 * C;                     // C even -> already even
            repack_pairs<<<((KtotPad << coL2) + 255) / 256, 256, 0, stream>>>(
                wb, wp2, coL2, coL2, KtotPad);
            if (C >= 32) {
                const int ctL2  = coL2 - 5;
                const int waves = (V / 32) << ctL2;         // MT=2, NT=2
                conv3d_wmma_cx<2, 2><<<waves / 8, 256, 0, stream>>>(
                    bufA, wp2, mlev[i], bufB, doL2, doL2, coL2, coL2, ctL2, 1);
            } else {  // block0: Cout=16 -> MT=2, NT=1
                const int waves = (V / 32);
                conv3d_wmma_cx<2, 1><<<waves / 8, 256, 0, stream>>>(
                    bufA, wp2, mlev[i], bufB, doL2, doL2, coL2, coL2, 0, 1);
            }
        }
        // ---- BN + ELU ----
        bn_stats_partial<<<gs, 256, 0, stream>>>(bufB, V, coL2, part);
        bn_stats_final<<<(2 * C + 255) / 256, 256, 0, stream>>>(part, 64, C, stats);
        bn_apply_elu<<<(int)((tot + 255) / 256), 256, 0, stream>>>(
            bufB, mlev[i], stats, cnt + i, gb, bb, tot, coL2);

        src = bufB;
        Cin = C;
    }

    // -------- 1x1 conv 512->1024 + masked global average pool ---------------
    final_reduce<<<4, 512, 0, stream>>>(bufB, mlev[5], Sbuf, nb);
    final_gemm<<<(4096 + 255) / 256, 256, 0, stream>>>(Sbuf, wfin, nb, (float*)d_out);
}